// TSConstruction_44169443672126
// MI455X (gfx1250) — compile-verified
//
#include <hip/hip_runtime.h>
#include <hip/hip_bf16.h>

// ---------------------------------------------------------------------------
// Problem constants
// ---------------------------------------------------------------------------
#define BB 32
#define NN 360
#define TFULL 512
#define TT 64
#define KSZ 8
#define MSEQ (BB * NN)           // 11520 sequences
#define ROWS ((size_t)MSEQ * 64) // 737280 flattened (m,t) rows
#define G3 24                    // 3*KS gates

typedef __attribute__((ext_vector_type(2))) float v2f;
typedef __attribute__((ext_vector_type(8))) float v8f;

__device__ __forceinline__ float sigmoidf_(float x) {
    return 1.0f / (1.0f + __expf(-x));
}

// f32 WMMA 16x16x4: D = A(16x4) * B(4x16) + C(16x16)
__device__ __forceinline__ v8f wmma4(v2f a, v2f b, v8f c) {
    return __builtin_amdgcn_wmma_f32_16x16x4_f32(
        /*neg_a=*/false, a, /*neg_b=*/false, b,
        /*c_mod=*/(short)0, c, /*reuse_a=*/false, /*reuse_b=*/false);
}

// ---------------------------------------------------------------------------
// Kernel W: xp[row, g] = bih[g] + sum_k X[row,k]*Wih[g,k]   (row-tiled WMMA)
// X: (ROWS, IN) row-major, Wih: (24, IN), XP: (ROWS, 24).
// Weights staged in LDS zero-padded to 32 gates -> unconditional B fragments,
// built once per wave; wave then grid-strides over 16-row tiles with a
// uniform trip count (EXEC all-ones through every WMMA).
// ---------------------------------------------------------------------------
template <int IN>
__global__ void __launch_bounds__(256)
xp_wmma_kernel(const float* __restrict__ X,
               const float* __restrict__ Wih,
               const float* __restrict__ bih,
               float* __restrict__ XP) {
    __shared__ float sW[32 * IN];  // [g][k], gates 24..31 zeroed
    __shared__ float sB[32];
    const int tid = threadIdx.x;
    for (int i = tid; i < 32 * IN; i += blockDim.x)
        sW[i] = (i < G3 * IN) ? Wih[i] : 0.0f;
    if (tid < 32) sB[tid] = (tid < G3) ? bih[tid] : 0.0f;
    __syncthreads();

    const int wave = tid >> 5;
    const int lane = tid & 31;
    const int hi   = lane >> 4;   // lane half selects K sub-pair
    const int col  = lane & 15;   // N index (tile-local), also A's M index
    const int g1   = col + 16;

    // Build B fragments + bias once (unconditional LDS reads)
    constexpr int NCH = IN / 4;
    v2f bA[NCH], bB[NCH];
#pragma unroll
    for (int c = 0; c < NCH; ++c) {
        const int k = c * 4 + 2 * hi;
        bA[c].x = sW[col * IN + k];
        bA[c].y = sW[col * IN + k + 1];
        bB[c].x = sW[g1 * IN + k];
        bB[c].y = sW[g1 * IN + k + 1];
    }
    const float b0v = sB[col];
    const float b1v = sB[g1];

    const size_t tiles  = ROWS / 16;  // 46080
    const size_t waveId = (size_t)blockIdx.x * (blockDim.x >> 5) + wave;
    const size_t nWaves = (size_t)gridDim.x * (blockDim.x >> 5);
    for (size_t tIdx = waveId; tIdx < tiles; tIdx += nWaves) {
        const size_t R0 = tIdx * 16;
        v8f c0, c1;
#pragma unroll
        for (int v = 0; v < 8; ++v) { c0[v] = b0v; c1[v] = b1v; }
        const size_t arow = R0 + (size_t)col;
#pragma unroll
        for (int c = 0; c < NCH; ++c) {
            const int k = c * 4 + 2 * hi;
            const v2f a = *(const v2f*)(X + arow * IN + k);  // 8B-aligned
            c0 = wmma4(a, bA[c], c0);
            c1 = wmma4(a, bB[c], c1);
        }
#pragma unroll
        for (int v = 0; v < 8; ++v) {
            const size_t r = R0 + (size_t)(v + 8 * hi);
            XP[r * G3 + col] = c0[v];
            if (g1 < G3) XP[r * G3 + g1] = c1[v];
        }
    }
}

// ---------------------------------------------------------------------------
// Kernel R: GRU recurrence, one thread per sequence. Reads precomputed xp.
// out layout: (MSEQ, 64, 16); direction d writes columns [8d, 8d+8).
// ---------------------------------------------------------------------------
__global__ void gru_rec_kernel(const float* __restrict__ XP,
                               const float* __restrict__ Whh,
                               const float* __restrict__ bhh,
                               float* __restrict__ out, int dir) {
    __shared__ float sW[G3 * 8];
    __shared__ float sB[G3];
    const int tid = threadIdx.x;
    for (int i = tid; i < G3 * 8; i += blockDim.x) sW[i] = Whh[i];
    if (tid < G3) sB[tid] = bhh[tid];
    __syncthreads();

    const size_t m = (size_t)blockIdx.x * blockDim.x + tid;  // < 11520
    float h[8];
#pragma unroll
    for (int j = 0; j < 8; ++j) h[j] = 0.0f;

    for (int s = 0; s < 64; ++s) {
        const int t = dir ? (63 - s) : s;
        const float* xg = XP + (m * 64 + (size_t)t) * G3;
        float gh[G3];
#pragma unroll
        for (int g = 0; g < G3; ++g) {
            float acc = sB[g];
#pragma unroll
            for (int j = 0; j < 8; ++j) acc += sW[g * 8 + j] * h[j];
            gh[g] = acc;
        }
        float hn[8];
#pragma unroll
        for (int j = 0; j < 8; ++j) {
            const float r  = sigmoidf_(xg[j]      + gh[j]);
            const float z  = sigmoidf_(xg[8 + j]  + gh[8 + j]);
            const float nv = tanhf(xg[16 + j] + r * gh[16 + j]);
            hn[j] = (1.0f - z) * nv + z * h[j];
        }
        float* o = out + (m * 64 + (size_t)t) * 16 + dir * 8;
#pragma unroll
        for (int j = 0; j < 8; ++j) { h[j] = hn[j]; o[j] = h[j]; }
    }
}

// ---------------------------------------------------------------------------
// Kernel F: feat = l2( leaky_relu( l1(last-step features) ) ), 8 per node.
// ---------------------------------------------------------------------------
__global__ void feat_kernel(const float* __restrict__ last,
                            const float* __restrict__ l1w, const float* __restrict__ l1b,
                            const float* __restrict__ l2w, const float* __restrict__ l2b,
                            float* __restrict__ feat) {
    const size_t m = (size_t)blockIdx.x * blockDim.x + threadIdx.x;  // < 11520
    const float* x = last + (m * 64 + 63) * 16;
    float f1[8];
#pragma unroll
    for (int o = 0; o < 8; ++o) {
        float a = l1b[o];
#pragma unroll
        for (int f = 0; f < 16; ++f) a += l1w[o * 16 + f] * x[f];
        f1[o] = (a >= 0.0f) ? a : 0.2f * a;
    }
    float* fo = feat + m * 8;
#pragma unroll
    for (int o = 0; o < 8; ++o) {
        float a = l2b[o];
#pragma unroll
        for (int f = 0; f < 8; ++f) a += l2w[o * 8 + f] * f1[f];
        fo[o] = a;
    }
}

// ---------------------------------------------------------------------------
// Threefry-2x32 (key=(0,42), 20 rounds) -> uniform [0,1) (JAX bit mapping)
// Classic counter scheme: pair (i, i+half) for flat index i.
// ---------------------------------------------------------------------------
__device__ __forceinline__ unsigned rotl32_(unsigned x, int r) {
    return (x << r) | (x >> (32 - r));
}
__device__ __forceinline__ float uniform_tf(unsigned flat) {
    const unsigned half = (unsigned)(BB * NN * NN) / 2u;  // 2073600
    unsigned c0, c1;
    const bool lo = flat < half;
    if (lo) { c0 = flat;        c1 = flat + half; }
    else    { c0 = flat - half; c1 = flat;        }
    const unsigned k0 = 0u, k1 = 42u;
    const unsigned k2 = 0x1BD11BDAu ^ k0 ^ k1;
    unsigned x0 = c0 + k0, x1 = c1 + k1;
#define TFR_(r) { x0 += x1; x1 = rotl32_(x1, r); x1 ^= x0; }
    TFR_(13) TFR_(15) TFR_(26) TFR_(6)  x0 += k1; x1 += k2 + 1u;
    TFR_(17) TFR_(29) TFR_(16) TFR_(24) x0 += k2; x1 += k0 + 2u;
    TFR_(13) TFR_(15) TFR_(26) TFR_(6)  x0 += k0; x1 += k1 + 3u;
    TFR_(17) TFR_(29) TFR_(16) TFR_(24) x0 += k1; x1 += k2 + 4u;
    TFR_(13) TFR_(15) TFR_(26) TFR_(6)  x0 += k2; x1 += k0 + 5u;
#undef TFR_
    const unsigned w = lo ? x0 : x1;
    return __uint_as_float((w >> 9) | 0x3F800000u) - 1.0f;
}

// ---------------------------------------------------------------------------
// Kernel L: logits = feat[b] @ feat[b]^T via WMMA, + gumbel, row-wise argmax.
// adj is exactly one-hot(argmax) with zeroed diagonal -> store only the index.
// One wave per (b, 16-row tile); sweeps 23 column tiles.
// ---------------------------------------------------------------------------
__global__ void adj_kernel(const float* __restrict__ feat, int* __restrict__ adjIdx) {
    const int b    = blockIdx.y;
    const int j0   = blockIdx.x * 16;
    const int lane = threadIdx.x;
    const int hi   = lane >> 4;
    const int col  = lane & 15;
    const float* fb = feat + (size_t)b * NN * 8;

    // A fragments (rows j0..j0+15, K=0..7 in two chunks)
    const int jm  = j0 + col;
    const bool jmv = jm < NN;
    const int ka  = 2 * hi;
    v2f a0, a1;
    a0.x = jmv ? fb[jm * 8 + ka]         : 0.0f;
    a0.y = jmv ? fb[jm * 8 + ka + 1]     : 0.0f;
    a1.x = jmv ? fb[jm * 8 + 4 + ka]     : 0.0f;
    a1.y = jmv ? fb[jm * 8 + 4 + ka + 1] : 0.0f;

    float best[8];
    int   bidx[8];
#pragma unroll
    for (int v = 0; v < 8; ++v) { best[v] = -3.402823466e38f; bidx[v] = 0; }

    for (int pt = 0; pt < 23; ++pt) {
        const int p = pt * 16 + col;
        const bool pv = p < NN;
        v2f b0, b1;
        b0.x = pv ? fb[p * 8 + ka]         : 0.0f;
        b0.y = pv ? fb[p * 8 + ka + 1]     : 0.0f;
        b1.x = pv ? fb[p * 8 + 4 + ka]     : 0.0f;
        b1.y = pv ? fb[p * 8 + 4 + ka + 1] : 0.0f;
        v8f c;
#pragma unroll
        for (int v = 0; v < 8; ++v) c[v] = 0.0f;
        c = wmma4(a0, b0, c);
        c = wmma4(a1, b1, c);
#pragma unroll
        for (int v = 0; v < 8; ++v) {
            const int j = j0 + v + 8 * hi;
            float val = -3.402823466e38f;
            if (pv && j < NN) {
                const unsigned flat = ((unsigned)(b * NN + j)) * (unsigned)NN + (unsigned)p;
                const float U = uniform_tf(flat);
                const float g = -logf(-logf(U + 1e-10f) + 1e-10f);
                val = c[v] + g;
            }
            if (val > best[v]) { best[v] = val; bidx[v] = p; }
        }
    }
    // reduce across the 16 lanes holding the same rows (xor within half)
#pragma unroll
    for (int off = 1; off < 16; off <<= 1) {
#pragma unroll
        for (int v = 0; v < 8; ++v) {
            const float ov = __shfl_xor(best[v], off, 32);
            const int   oi = __shfl_xor(bidx[v], off, 32);
            if (ov > best[v] || (ov == best[v] && oi < bidx[v])) {
                best[v] = ov; bidx[v] = oi;
            }
        }
    }
    if (col == 0) {
#pragma unroll
        for (int v = 0; v < 8; ++v) {
            const int j = j0 + v + 8 * hi;
            if (j < NN) adjIdx[b * NN + j] = bidx[v];
        }
    }
}

// ---------------------------------------------------------------------------
// Kernel E: DCGRU encoder (64 steps) + decoder (64 steps).
// One block per batch; node state in LDS; adj@x is a gather through adjIdx
// (one-hot rows, diagonal hits -> zero).
// ---------------------------------------------------------------------------
__global__ void __launch_bounds__(384)
encdec_kernel(const float* __restrict__ rs, const int* __restrict__ adjIdx,
              const float* __restrict__ egw, const float* __restrict__ egb,
              const float* __restrict__ ecw, const float* __restrict__ ecb,
              const float* __restrict__ dgw, const float* __restrict__ dgb,
              const float* __restrict__ dcw, const float* __restrict__ dcb,
              const float* __restrict__ pw,  const float* __restrict__ pb,
              float* __restrict__ out) {
    const int b = blockIdx.x;
    const int n = threadIdx.x;
    __shared__ float h[3][NN];
    __shared__ float x0a[NN], x0b[NN], x1a[NN], x1b[NN];
    __shared__ int   idx[NN];
    __shared__ float W[2][3][21];  // [enc/dec][layer][gw(12) gb(2) cw(6) cb(1)]

    if (n == 0) {
        for (int l = 0; l < 3; ++l) {
            for (int f = 0; f < 12; ++f) { W[0][l][f] = egw[l * 12 + f]; W[1][l][f] = dgw[l * 12 + f]; }
            for (int k = 0; k < 2;  ++k) { W[0][l][12 + k] = egb[l * 2 + k]; W[1][l][12 + k] = dgb[l * 2 + k]; }
            for (int f = 0; f < 6;  ++f) { W[0][l][14 + f] = ecw[l * 6 + f]; W[1][l][14 + f] = dcw[l * 6 + f]; }
            W[0][l][20] = ecb[l]; W[1][l][20] = dcb[l];
        }
    }
    for (int i = n; i < NN; i += blockDim.x) {
        idx[i] = adjIdx[b * NN + i];
        h[0][i] = 0.0f; h[1][i] = 0.0f; h[2][i] = 0.0f;
    }
    __syncthreads();

    const bool  act   = n < NN;
    const float projw = pw[0];
    const float projb = pb[0];

    // One DCGRU cell (all threads call; barriers are uniform)
    auto cell = [&](float o_n, float h_n, const float* Wl) -> float {
        float r = 0.0f, u = 0.0f;
        float x1x = 0.0f, x1y = 0.0f;
        // ---- gate gconv: x0 = (o, h)
        if (act) { x0a[n] = o_n; x0b[n] = h_n; }
        __syncthreads();
        if (act) {
            const int j = idx[n];
            if (j != n) { x1x = x0a[j]; x1y = x0b[j]; }
            x1a[n] = x1x; x1b[n] = x1y;
        }
        __syncthreads();
        if (act) {
            const int j = idx[n];
            float gx = 0.0f, gy = 0.0f;
            if (j != n) { gx = x1a[j]; gy = x1b[j]; }
            const float x2x = 2.0f * gx - o_n;
            const float x2y = 2.0f * gy - h_n;
            // xc = [o, x1x, x2x, h, x1y, x2y]
            const float s0 = Wl[0] * o_n + Wl[2] * x1x + Wl[4] * x2x +
                             Wl[6] * h_n + Wl[8] * x1y + Wl[10] * x2y + Wl[12];
            const float s1 = Wl[1] * o_n + Wl[3] * x1x + Wl[5] * x2x +
                             Wl[7] * h_n + Wl[9] * x1y + Wl[11] * x2y + Wl[13];
            r = sigmoidf_(s0);
            u = sigmoidf_(s1);
        }
        __syncthreads();  // done reading x0/x1 buffers
        // ---- candidate gconv: x0 = (o, r*h)
        const float st2 = act ? r * h_n : 0.0f;
        if (act) { x0a[n] = o_n; x0b[n] = st2; }
        __syncthreads();
        float y1x = 0.0f, y1y = 0.0f;
        if (act) {
            const int j = idx[n];
            if (j != n) { y1x = x0a[j]; y1y = x0b[j]; }
            x1a[n] = y1x; x1b[n] = y1y;
        }
        __syncthreads();
        float hnew = 0.0f;
        if (act) {
            const int j = idx[n];
            float gx = 0.0f, gy = 0.0f;
            if (j != n) { gx = x1a[j]; gy = x1b[j]; }
            const float y2x = 2.0f * gx - o_n;
            const float y2y = 2.0f * gy - st2;
            const float cpre = Wl[14] * o_n + Wl[15] * y1x + Wl[16] * y2x +
                               Wl[17] * st2 + Wl[18] * y1y + Wl[19] * y2y + Wl[20];
            const float c = tanhf(cpre);
            hnew = u * h_n + (1.0f - u) * c;
        }
        __syncthreads();  // buffers free for next call
        return hnew;
    };

    // ---- encoder ----
    for (int t = 0; t < TT; ++t) {
        float o = act ? rs[((size_t)b * NN + n) * TT + t] : 0.0f;
        for (int l = 0; l < 3; ++l) {
            const float hn_ = cell(o, act ? h[l][n] : 0.0f, W[0][l]);
            if (act) h[l][n] = hn_;
            o = hn_;
        }
    }
    // ---- decoder ----
    float o = 0.0f;
    for (int t = 0; t < TT; ++t) {
        for (int l = 0; l < 3; ++l) {
            const float hn_ = cell(o, act ? h[l][n] : 0.0f, W[1][l]);
            if (act) h[l][n] = hn_;
            o = hn_;
        }
        o = o * projw + projb;
        if (act) out[((size_t)b * NN + n) * TT + t] = o;
    }
}

// ---------------------------------------------------------------------------
// Host-side launcher
// ---------------------------------------------------------------------------
extern "C" void kernel_launch(void* const* d_in, const int* in_sizes, int n_in,
                              void* d_out, int out_size, void* d_ws, size_t ws_size,
                              hipStream_t stream) {
    const float* full_seq = (const float*)d_in[0];
    const float* rs       = (const float*)d_in[1];
    // d_in[2] node_feas, d_in[3] temperature: unused by the forward pass
    const float* g0wih = (const float*)d_in[4];
    const float* g0whh = (const float*)d_in[5];
    const float* g0bih = (const float*)d_in[6];
    const float* g0bhh = (const float*)d_in[7];
    const float* gwih  = (const float*)d_in[8];
    const float* gwhh  = (const float*)d_in[9];
    const float* gbih  = (const float*)d_in[10];
    const float* gbhh  = (const float*)d_in[11];
    const float* l1w   = (const float*)d_in[12];
    const float* l1b   = (const float*)d_in[13];
    const float* l2w   = (const float*)d_in[14];
    const float* l2b   = (const float*)d_in[15];
    const float* egw   = (const float*)d_in[16];
    const float* egb   = (const float*)d_in[17];
    const float* ecw   = (const float*)d_in[18];
    const float* ecb   = (const float*)d_in[19];
    const float* dgw   = (const float*)d_in[20];
    const float* dgb   = (const float*)d_in[21];
    const float* dcw   = (const float*)d_in[22];
    const float* dcb   = (const float*)d_in[23];
    const float* pw    = (const float*)d_in[24];
    const float* pb    = (const float*)d_in[25];

    float* ws = (float*)d_ws;
    const size_t BUF = ROWS * 16;       // 11,796,480 floats per ping-pong buf
    float* bufA = ws;
    float* bufB = ws + BUF;
    float* XP   = ws + 2 * BUF;         // ROWS * 24 floats
    float* feat = XP + ROWS * 24;       // MSEQ * 8 floats
    int* adjIdx = (int*)(feat + (size_t)MSEQ * 8);

    const float* layerIn = full_seq;    // (ROWS, 8) view of full_seq
    float* layerOut = bufA;
    for (int layer = 0; layer < 4; ++layer) {
        for (int d = 0; d < 2; ++d) {
            const float *wih, *whh, *bih, *bhh;
            if (layer == 0) {
                wih = g0wih + d * G3 * 8;  whh = g0whh + d * G3 * 8;
                bih = g0bih + d * G3;      bhh = g0bhh + d * G3;
            } else {
                const int l = layer - 1;
                wih = gwih + (size_t)(l * 2 + d) * G3 * 16;
                whh = gwhh + (size_t)(l * 2 + d) * G3 * 8;
                bih = gbih + (size_t)(l * 2 + d) * G3;
                bhh = gbhh + (size_t)(l * 2 + d) * G3;
            }
            if (layer == 0) {
                xp_wmma_kernel<8><<<dim3(720), dim3(256), 0, stream>>>(layerIn, wih, bih, XP);
            } else {
                xp_wmma_kernel<16><<<dim3(720), dim3(256), 0, stream>>>(layerIn, wih, bih, XP);
            }
            gru_rec_kernel<<<dim3(45), dim3(256), 0, stream>>>(XP, whh, bhh, layerOut, d);
        }
        layerIn  = layerOut;
        layerOut = (layerOut == bufA) ? bufB : bufA;
    }

    feat_kernel<<<dim3(45), dim3(256), 0, stream>>>(layerIn, l1w, l1b, l2w, l2b, feat);
    adj_kernel<<<dim3(23, 32), dim3(32), 0, stream>>>(feat, adjIdx);
    encdec_kernel<<<dim3(32), dim3(384), 0, stream>>>(rs, adjIdx, egw, egb, ecw, ecb,
                                                      dgw, dgb, dcw, dcb, pw, pb,
                                                      (float*)d_out);
}